// Model_69578470195328
// MI455X (gfx1250) — compile-verified
//
#include <hip/hip_runtime.h>
#include <hip/hip_bf16.h>

// Fused decoupled-GCN block for MI455X (gfx1250, wave32, WMMA + TDM).
//   stage1: y = BN(X @ W + lb)          via v_wmma_f32_16x16x4_f32
//   stage2: y2 = sum_k,v y * normA      via v_wmma_f32_16x16x4_f32 (acc over k)
//   epilog: out = relu((y2 + x)*s2 + o2)
// One workgroup = one (n, 4-t) slab; intermediates live in LDS (~129 KB) so
// HBM traffic is read-x + write-out only (~160 MB -> ~7 us floor @ 23.3 TB/s).
// X slab and W are DMA'd into LDS by the Tensor Data Mover; TDM's LDS-pad
// feature (+2 DWORDs every 64) creates the 66-float row stride in hardware.

typedef float v2f __attribute__((ext_vector_type(2)));
typedef float v8f __attribute__((ext_vector_type(8)));
typedef unsigned int u32x4 __attribute__((ext_vector_type(4)));
typedef int i32x4 __attribute__((ext_vector_type(4)));
typedef int i32x8 __attribute__((ext_vector_type(8)));

#ifndef __has_builtin
#define __has_builtin(x) 0
#endif
#if defined(__gfx1250__) && __has_builtin(__builtin_amdgcn_tensor_load_to_lds) && \
    __has_builtin(__builtin_amdgcn_s_wait_tensorcnt)
#define USE_TDM 1
#else
#define USE_TDM 0
#endif

namespace {
constexpr int C      = 64;    // d_model
constexpr int DE     = 192;   // expanded channels (3*C)
constexpr int G      = 8;     // adjacency groups
constexpr int K3     = 3;     // SUB
constexpr int NV     = 19;    // joints
constexpr int VP     = 20;    // padded joints (adjacency K-pad)
constexpr int T_ALL  = 32;
constexpr int TB     = 4;     // t per workgroup
constexpr int ROWS   = 80;    // 76 (t,v) rows + 4 zero pad rows
constexpr int VROWS  = TB * NV;   // 76 valid rows, contiguous t*19+v
constexpr int XS     = 66;    // sX row stride (floats) == TDM pad stride
constexpr int YS     = 66;    // sY row stride
constexpr int BLOCK  = 256;   // 8 waves

// dynamic-LDS float offsets
constexpr int OFF_W  = 0;                    // [64][192]
constexpr int OFF_X  = OFF_W + C * DE;       // [80][XS]
constexpr int OFF_Y  = OFF_X + ROWS * XS;    // [80][YS]
constexpr int OFF_A  = OFF_Y + ROWS * YS;    // [3][8][20][20] normalized adjacency
constexpr int OFF_S1 = OFF_A + K3 * G * VP * VP;
constexpr int OFF_O1 = OFF_S1 + DE;
constexpr int OFF_S2 = OFF_O1 + DE;
constexpr int OFF_O2 = OFF_S2 + C;
constexpr int LDS_FLOATS = OFF_O2 + C;       // 32960 floats = 131840 B
}

__device__ __forceinline__ v8f wmma4(v2f a, v2f b, v8f c) {
  // D = A(16x4 f32) * B(4x16 f32) + C(16x16 f32), full wave32 op
  return __builtin_amdgcn_wmma_f32_16x16x4_f32(
      /*neg_a=*/false, a, /*neg_b=*/false, b,
      /*c_mod=*/(short)0, c, /*reuse_a=*/false, /*reuse_b=*/false);
}

#if USE_TDM
// Issue one 2-D TDM load: dim1 rows of dim0 elements (4B), optional LDS pad.
// Descriptor layout per CDNA5 ISA.
__device__ __forceinline__ void tdm_load_2d(uint32_t lds_addr, const void* gptr,
                                            uint32_t dim0, uint32_t dim1,
                                            uint32_t stride0, uint32_t tile0,
                                            uint32_t tile1, uint32_t g1w0_pad) {
  uint64_t ga = (uint64_t)(uintptr_t)gptr;
  u32x4 g0;
  g0[0] = 1u;                                   // count=1, user descriptor
  g0[1] = lds_addr;                             // LDS byte address
  g0[2] = (uint32_t)ga;                         // global_addr[31:0]
  g0[3] = (uint32_t)(ga >> 32) | (2u << 30);    // global_addr[56:32] | type=2
  i32x8 g1;
  g1[0] = (int)((2u << 16) | g1w0_pad);         // data_size=4B | pad config
  g1[1] = (int)(dim0 << 16);                    // tensor_dim0[15:0]
  g1[2] = (int)((dim0 >> 16) | (dim1 << 16));   // tensor_dim0[31:16]|dim1[15:0]
  g1[3] = (int)((dim1 >> 16) | (tile0 << 16));  // tensor_dim1[31:16]|tile_dim0
  g1[4] = (int)(tile1 & 0xFFFFu);               // tile_dim1 | tile_dim2=0
  g1[5] = (int)stride0;                         // tensor_dim0_stride[31:0]
  g1[6] = 0;                                    // stride0 hi | dim1_stride lo
  g1[7] = 0;
  i32x4 z4 = {0, 0, 0, 0};
#if __clang_major__ >= 23
  i32x8 z8 = {0, 0, 0, 0, 0, 0, 0, 0};
  __builtin_amdgcn_tensor_load_to_lds(g0, g1, z4, z4, z8, 0);
#else
  __builtin_amdgcn_tensor_load_to_lds(g0, g1, z4, z4, 0);
#endif
}
#endif

__global__ __launch_bounds__(BLOCK) void dgcn_fused(
    const float* __restrict__ xg,   const float* __restrict__ para,
    const float* __restrict__ Wg,   const float* __restrict__ lb,
    const float* __restrict__ g1,   const float* __restrict__ b1,
    const float* __restrict__ m1,   const float* __restrict__ v1,
    const float* __restrict__ g2,   const float* __restrict__ b2,
    const float* __restrict__ m2,   const float* __restrict__ v2,
    float* __restrict__ out)
{
  extern __shared__ float smem[];
  float* sW  = smem + OFF_W;
  float* sX  = smem + OFF_X;
  float* sY  = smem + OFF_Y;
  float* sA  = smem + OFF_A;
  float* sS1 = smem + OFF_S1;
  float* sO1 = smem + OFF_O1;
  float* sS2 = smem + OFF_S2;
  float* sO2 = smem + OFF_O2;

  const int tid  = threadIdx.x;
  const int lane = tid & 31;
  const int wave = tid >> 5;         // 0..7
  const int hi   = lane >> 4;        // 0/1 half-wave (K select in WMMA frags)
  const int ln16 = lane & 15;

  const int n  = blockIdx.y;
  const int t0 = blockIdx.x * TB;
  const size_t xbase = ((size_t)n * T_ALL + t0) * (size_t)(NV * C);

#if USE_TDM
  if (wave == 0) {
    // X slab: 76 rows x 64 floats contiguous in HBM -> LDS rows of stride 66
    // via TDM pad: +2 DWORDs (pad_amount enc 1) every 64 DWORDs (interval enc 5)
    tdm_load_2d((uint32_t)(uintptr_t)sX, xg + xbase,
                /*dim0=*/C, /*dim1=*/VROWS, /*stride0=*/C,
                /*tile0=*/C, /*tile1=*/VROWS,
                /*pad=*/(1u << 20) | (5u << 22) | (1u << 25));
    // W: 12288 contiguous floats, no pad
    tdm_load_2d((uint32_t)(uintptr_t)sW, Wg,
                /*dim0=*/C * DE, /*dim1=*/1, /*stride0=*/C * DE,
                /*tile0=*/C * DE, /*tile1=*/1, /*pad=*/0u);
  }
#else
  {  // fallback: cooperative vector loads
    const float4* Wg4 = (const float4*)Wg;
    float4* sW4 = (float4*)sW;
    for (int i = tid; i < C * DE / 4; i += BLOCK) sW4[i] = Wg4[i];
    for (int i = tid; i < VROWS * C; i += BLOCK) {
      int row = i >> 6, c = i & 63;
      sX[row * XS + c] = xg[xbase + i];
    }
  }
#endif

  // zero the 4 trailing pad rows of sX (rows 76..79) so stage1 stays finite
  for (int i = tid; i < (ROWS - VROWS) * C; i += BLOCK)
    sX[(VROWS + (i >> 6)) * XS + (i & 63)] = 0.0f;

  if (tid < DE) {  // BN-after-linear folded: y*s1 + o1
    float inv = g1[tid] * rsqrtf(v1[tid] + 1e-5f);
    sS1[tid] = inv;
    sO1[tid] = lb[tid] * inv + b1[tid] - m1[tid] * inv;
  }
  if (tid < C) {   // BN1 folded
    float inv = g2[tid] * rsqrtf(v2[tid] + 1e-5f);
    sS2[tid] = inv;
    sO2[tid] = b2[tid] - m2[tid] * inv;
  }
  for (int i = tid; i < K3 * G * VP; i += BLOCK) {  // row-normalized adjacency
    int v = i % VP, kg = i / VP;          // kg = k*8+g
    float* dst = sA + (kg * VP + v) * VP;
    if (v < NV) {
      const float* src = para + (kg * NV + v) * NV;
      float ss = 0.0f;
      for (int w = 0; w < NV; ++w) { float p = src[w]; ss += p * p; }
      float inv = 1.0f / (sqrtf(ss) + 1e-4f);
      for (int w = 0; w < NV; ++w) dst[w] = src[w] * inv;
      dst[NV] = 0.0f;
    } else {
      for (int w = 0; w < VP; ++w) dst[w] = 0.0f;  // K-pad row MUST be 0:
    }                                              // it kills the v=19 products
  }

#if USE_TDM
  if (wave == 0) __builtin_amdgcn_s_wait_tensorcnt(0);  // TDM done -> LDS valid
#endif
  __syncthreads();

  const v8f vzero = {0.f, 0.f, 0.f, 0.f, 0.f, 0.f, 0.f, 0.f};
  v8f acc2[4];
  #pragma unroll
  for (int i = 0; i < 4; ++i) acc2[i] = vzero;

  // ---- k-slice loop: stage1 (64 channels) then stage2 (accumulate) ------
  for (int kb = 0; kb < K3; ++kb) {
    // stage 1: sY[80 x 64] = BN(sX[80x64] @ sW[:, kb*64 : kb*64+64])
    for (int tile = wave; tile < 20; tile += 8) {
      int mt = tile >> 2, nt = tile & 3;
      int m = mt * 16 + ln16;
      int dcol = kb * C + nt * 16 + ln16;
      const float* xrow = sX + m * XS;
      const float* wcol = sW + dcol;
      v8f acc = vzero;
      #pragma unroll
      for (int kk = 0; kk < C; kk += 4) {
        v2f a = *(const v2f*)(xrow + kk + 2 * hi);         // A[m][k0..k0+1]
        v2f b;
        b.x = wcol[(kk + 2 * hi) * DE];                    // B[k][n]
        b.y = wcol[(kk + 2 * hi + 1) * DE];
        acc = wmma4(a, b, acc);
      }
      float s = sS1[dcol], o = sO1[dcol];
      int ncl = nt * 16 + ln16;
      #pragma unroll
      for (int j = 0; j < 8; ++j) {
        int row = mt * 16 + j + 8 * hi;
        sY[row * YS + ncl] = acc[j] * s + o;
      }
    }
    __syncthreads();

    // stage 2: per group g==wave, M=(8 channels x 4 t)=32, K=v(pad 20), N=w(pad 32)
    {
      const int gg = wave;
      const float* Ab = sA + (kb * G + gg) * VP * VP;
      #pragma unroll
      for (int mt = 0; mt < 2; ++mt) {
        #pragma unroll
        for (int nt = 0; nt < 2; ++nt) {
          int m  = mt * 16 + ln16;
          int c  = gg + 8 * (m >> 2);
          int tl = m & 3;
          int w  = nt * 16 + ln16;
          bool wok = (w < NV);
          int wc = wok ? w : 0;           // clamp so LDS loads stay in-bounds
          v8f acc = acc2[mt * 2 + nt];
          #pragma unroll
          for (int vv = 0; vv < 5; ++vv) {
            int vb = vv * 4 + 2 * hi;     // v in 0..19; v==19 row of A is 0
            v2f a, b;
            a.x = sY[(tl * NV + vb) * YS + c];
            a.y = sY[(tl * NV + vb + 1) * YS + c];
            b.x = wok ? Ab[vb * VP + wc] : 0.0f;
            b.y = wok ? Ab[(vb + 1) * VP + wc] : 0.0f;
            acc = wmma4(a, b, acc);
          }
          acc2[mt * 2 + nt] = acc;
        }
      }
    }
    __syncthreads();
  }

  // ---- spill y2 fragments to LDS (reuse sY) for coalesced epilogue ------
  {
    const int gg = wave;
    #pragma unroll
    for (int mt = 0; mt < 2; ++mt) {
      #pragma unroll
      for (int nt = 0; nt < 2; ++nt) {
        v8f acc = acc2[mt * 2 + nt];
        int w = nt * 16 + ln16;
        if (w < NV) {
          #pragma unroll
          for (int j = 0; j < 8; ++j) {
            int m  = mt * 16 + j + 8 * hi;
            int c  = gg + 8 * (m >> 2);
            int tl = m & 3;
            sY[(tl * NV + w) * YS + c] = acc[j];
          }
        }
      }
    }
  }
  __syncthreads();

  // ---- epilogue: out = relu((y2 + x)*s2 + o2), fully coalesced ----------
  for (int i = tid; i < VROWS * C; i += BLOCK) {
    int row = i >> 6, c = i & 63;
    float z  = sY[row * YS + c];
    float xr = sX[row * XS + c];
    float o  = (z + xr) * sS2[c] + sO2[c];
    out[xbase + i] = fmaxf(o, 0.0f);
  }
}

extern "C" void kernel_launch(void* const* d_in, const int* in_sizes, int n_in,
                              void* d_out, int out_size, void* d_ws, size_t ws_size,
                              hipStream_t stream) {
  const float* x    = (const float*)d_in[0];
  const float* para = (const float*)d_in[1];
  const float* W    = (const float*)d_in[2];
  const float* lb   = (const float*)d_in[3];
  const float* g1   = (const float*)d_in[4];
  const float* b1   = (const float*)d_in[5];
  const float* m1   = (const float*)d_in[6];
  const float* v1   = (const float*)d_in[7];
  const float* g2   = (const float*)d_in[8];
  const float* b2   = (const float*)d_in[9];
  const float* m2   = (const float*)d_in[10];
  const float* v2   = (const float*)d_in[11];
  float* out = (float*)d_out;

  const int N = in_sizes[0] / (T_ALL * NV * C);   // 512
  dim3 grid(T_ALL / TB, N);                       // (8, 512)
  size_t shmem = (size_t)LDS_FLOATS * sizeof(float);  // ~129 KB of 320 KB WGP LDS
  dgcn_fused<<<grid, BLOCK, shmem, stream>>>(x, para, W, lb, g1, b1, m1, v1,
                                             g2, b2, m2, v2, out);
}